// FixedSizeActorPool_62508954026545
// MI455X (gfx1250) — compile-verified
//
#include <hip/hip_runtime.h>
#include <hip/hip_bf16.h>

typedef float v2f __attribute__((ext_vector_type(2)));
typedef float v8f __attribute__((ext_vector_type(8)));
typedef float f4  __attribute__((ext_vector_type(4)));

#define BATCH 2048
#define CAST  128
#define HID   256
#define INF_  256   // input features

__device__ __forceinline__ float sigmoidf(float v) {
    return 1.0f / (1.0f + __expf(-v));
}

// ---------------------------------------------------------------------------
// Kernel 1: fused GRU cell via V_WMMA_F32_16X16X4_F32.
// One wave computes a 16(batch) x 16(hidden) tile of new_selected.
// Six f32 accumulators: {r,z,n} x {x@w_ih^T, h@w_hh^T}.
// A-frag 16x4 layout: lane = m + 16*(k>>1), vgpr = k&1  -> contiguous b64 load.
// B-frag 4x16 layout: lane = n + 16*(k>>1), vgpr = k&1  -> contiguous b64 load
//                     of transposed weight row (w[j0+n, k..k+1]).
// ---------------------------------------------------------------------------
__global__ void __launch_bounds__(256)
gru_wmma_kernel(const float* __restrict__ x,
                const long long* __restrict__ batch_idxs,
                const long long* __restrict__ actor_ids,
                const float* __restrict__ state,
                const float* __restrict__ w_ih,
                const float* __restrict__ w_hh,
                const float* __restrict__ b_ih,
                const float* __restrict__ b_hh,
                float* __restrict__ new_sel)
{
    const int wid    = (blockIdx.x << 3) + (threadIdx.x >> 5); // 8 waves/block
    const int lane   = threadIdx.x & 31;
    const int m_tile = wid >> 4;          // 128 batch tiles
    const int n_tile = wid & 15;          // 16 hidden tiles
    const int m0     = m_tile << 4;
    const int j0     = n_tile << 4;
    const int half   = lane >> 4;         // 0..1
    const int l15    = lane & 15;

    // A-fragment source rows for this lane (row m, k-pair base 2*half)
    const int m    = m0 + l15;
    int bidx = (int)batch_idxs[m];
    int aidm = (int)actor_ids[bidx];
    aidm = aidm < 0 ? 0 : (aidm > CAST - 1 ? CAST - 1 : aidm);

    const float* xA = x     + (size_t)m * INF_ + 2 * half;
    const float* hA = state + ((size_t)bidx * CAST + aidm) * HID + 2 * half;

    // B-fragment source rows (weight row j0+l15 per gate, transposed access)
    const float* wr = w_ih + (size_t)(0 * HID + j0 + l15) * INF_ + 2 * half;
    const float* wz = w_ih + (size_t)(1 * HID + j0 + l15) * INF_ + 2 * half;
    const float* wn = w_ih + (size_t)(2 * HID + j0 + l15) * INF_ + 2 * half;
    const float* ur = w_hh + (size_t)(0 * HID + j0 + l15) * HID  + 2 * half;
    const float* uz = w_hh + (size_t)(1 * HID + j0 + l15) * HID  + 2 * half;
    const float* un = w_hh + (size_t)(2 * HID + j0 + l15) * HID  + 2 * half;

    v8f acc_ir = {}, acc_hr = {};
    v8f acc_iz = {}, acc_hz = {};
    v8f acc_in = {}, acc_hn = {};

    #pragma unroll 4
    for (int kk = 0; kk < INF_ / 4; ++kk) {
        const int kb = kk * 4;
        v2f ax = *(const v2f*)(xA + kb);
        v2f ah = *(const v2f*)(hA + kb);
        v2f br = *(const v2f*)(wr + kb);
        v2f bz = *(const v2f*)(wz + kb);
        v2f bn = *(const v2f*)(wn + kb);
        v2f cr = *(const v2f*)(ur + kb);
        v2f cz = *(const v2f*)(uz + kb);
        v2f cn = *(const v2f*)(un + kb);

        acc_ir = __builtin_amdgcn_wmma_f32_16x16x4_f32(false, ax, false, br,
                                                       (short)0, acc_ir, false, false);
        acc_hr = __builtin_amdgcn_wmma_f32_16x16x4_f32(false, ah, false, cr,
                                                       (short)0, acc_hr, false, false);
        acc_iz = __builtin_amdgcn_wmma_f32_16x16x4_f32(false, ax, false, bz,
                                                       (short)0, acc_iz, false, false);
        acc_hz = __builtin_amdgcn_wmma_f32_16x16x4_f32(false, ah, false, cz,
                                                       (short)0, acc_hz, false, false);
        acc_in = __builtin_amdgcn_wmma_f32_16x16x4_f32(false, ax, false, bn,
                                                       (short)0, acc_in, false, false);
        acc_hn = __builtin_amdgcn_wmma_f32_16x16x4_f32(false, ah, false, cn,
                                                       (short)0, acc_hn, false, false);
    }

    // Epilogue. C/D 16x16 layout: element (m,n): vgpr v = m&7, lane = n + 16*(m>>3)
    const int j = j0 + l15;                 // output hidden index for this lane
    const float bir = b_ih[j],           bhr = b_hh[j];
    const float biz = b_ih[HID + j],     bhz = b_hh[HID + j];
    const float bin = b_ih[2 * HID + j], bhn = b_hh[2 * HID + j];

    #pragma unroll
    for (int v = 0; v < 8; ++v) {
        const int mv = m0 + v + 8 * half;
        int bv = (int)batch_idxs[mv];
        int av = (int)actor_ids[bv];
        av = av < 0 ? 0 : (av > CAST - 1 ? CAST - 1 : av);
        const float h_old = state[((size_t)bv * CAST + av) * HID + j];

        const float r  = sigmoidf((acc_ir[v] + bir) + (acc_hr[v] + bhr));
        const float z  = sigmoidf((acc_iz[v] + biz) + (acc_hz[v] + bhz));
        const float nn = tanhf((acc_in[v] + bin) + r * (acc_hn[v] + bhn));
        new_sel[(size_t)mv * HID + j] = (1.0f - z) * nn + z * h_old;
    }
}

// ---------------------------------------------------------------------------
// Kernel 2: stream state -> new_state (268 MB) with non-temporal float4 ops,
// substituting the GRU output row and zeroing story-stop slabs.
// One block per batch row; 128x256 floats = 8192 float4 per slab.
// ---------------------------------------------------------------------------
__global__ void __launch_bounds__(256)
state_update_kernel(const float* __restrict__ state,
                    const long long* __restrict__ actor_ids,
                    const long long* __restrict__ stop_idxs,
                    const float* __restrict__ new_sel,
                    float* __restrict__ out_state)
{
    const int b = blockIdx.x;
    __shared__ int s_stop;
    __shared__ int s_aid;

    if (threadIdx.x == 0) {
        s_stop = 0;
        int a = (int)actor_ids[b];
        s_aid = a < 0 ? 0 : (a > CAST - 1 ? CAST - 1 : a);
    }
    __syncthreads();
    if (threadIdx.x < 128) {
        if ((long long)b == stop_idxs[threadIdx.x]) s_stop = 1;
    }
    __syncthreads();

    const bool stop = (s_stop != 0);
    const int  aid  = s_aid;

    const f4* __restrict__ src = (const f4*)(state     + (size_t)b * CAST * HID);
    const f4* __restrict__ sel = (const f4*)(new_sel   + (size_t)b * HID);
    f4*       __restrict__ dst = (f4*)(out_state       + (size_t)b * CAST * HID);

    const f4 zero = {0.f, 0.f, 0.f, 0.f};
    for (int i = threadIdx.x; i < CAST * HID / 4; i += 256) {
        f4 val;
        if (stop) {
            val = zero;
        } else {
            const int c = i >> 6;                 // 64 float4 per 256-float row
            if (c == aid) val = sel[i & 63];
            else          val = __builtin_nontemporal_load(src + i);
        }
        __builtin_nontemporal_store(val, dst + i);
    }
}

extern "C" void kernel_launch(void* const* d_in, const int* in_sizes, int n_in,
                              void* d_out, int out_size, void* d_ws, size_t ws_size,
                              hipStream_t stream) {
    const float*     x          = (const float*)d_in[0];
    const long long* batch_idxs = (const long long*)d_in[1];
    const long long* actor_ids  = (const long long*)d_in[2];
    const long long* stop_idxs  = (const long long*)d_in[3];
    const float*     state      = (const float*)d_in[4];
    const float*     w_ih       = (const float*)d_in[5];
    const float*     w_hh       = (const float*)d_in[6];
    const float*     b_ih       = (const float*)d_in[7];
    const float*     b_hh       = (const float*)d_in[8];

    float* new_sel   = (float*)d_out;                        // (2048, 256)
    float* out_state = (float*)d_out + (size_t)BATCH * HID;  // (2048, 128, 256)

    // 2048 waves total (128 batch-tiles x 16 hidden-tiles), 8 waves per block
    gru_wmma_kernel<<<BATCH * 16 / 16 / 8, 256, 0, stream>>>(
        x, batch_idxs, actor_ids, state, w_ih, w_hh, b_ih, b_hh, new_sel);

    state_update_kernel<<<BATCH, 256, 0, stream>>>(
        state, actor_ids, stop_idxs, new_sel, out_state);
}